// AttMemoryLayer_85495618994942
// MI455X (gfx1250) — compile-verified
//
#include <hip/hip_runtime.h>

typedef float v2f __attribute__((ext_vector_type(2)));
typedef float v8f __attribute__((ext_vector_type(8)));

#define SLEN 4096
#define DDIM 256
#define NWAVES 16
#define BLOCK (NWAVES * 32)

__global__ __launch_bounds__(BLOCK)
void AttMemoryLayer_kernel(const float* __restrict__ aspect,
                           const float* __restrict__ memory,
                           const float* __restrict__ W,
                           const float* __restrict__ bias,
                           float* __restrict__ out) {
    __shared__ float g[SLEN];      // logits -> exp -> alfa (in place)
    __shared__ float wm[DDIM];     // Wm = W[:256]
    __shared__ float red[BLOCK];   // reduction scratch
    __shared__ float s_ab;

    const int tid  = threadIdx.x;
    const int lane = tid & 31;
    const int wave = tid >> 5;
    const int b    = blockIdx.x;
    const float* mem = memory + (size_t)b * SLEN * DDIM;

    // ---- Phase A: preload Wm; a_b = dot(aspect[b], Wa) + bias ----
    float part = 0.0f;
    if (tid < DDIM) {
        wm[tid] = W[tid];
        part = aspect[b * DDIM + tid] * W[DDIM + tid];
    }
    red[tid] = part;
    __syncthreads();
    for (int off = BLOCK / 2; off > 0; off >>= 1) {
        if (tid < off) red[tid] += red[tid + off];
        __syncthreads();
    }
    if (tid == 0) s_ab = red[0] + bias[0];
    __syncthreads();
    const float ab = s_ab;
    __syncthreads();

    // ---- Phase B: g[s] = tanh(dot(mem[s,:], Wm) + ab) via WMMA f32 16x16x4 ----
    // A (16x4 f32): M = lane&15 ; VGPR0 holds K = 0 (lanes 0-15) / K = 2 (lanes 16-31)
    const int mrow = lane & 15;
    const int koff = (lane >> 4) << 1;           // 0 or 2
    for (int tile = wave; tile < SLEN / 16; tile += NWAVES) {
        const int r0 = tile * 16;
        v8f acc = {};
        const float* rowp = mem + (size_t)(r0 + mrow) * DDIM + koff;
        for (int k0 = 0; k0 < DDIM; k0 += 4) {
            v2f a;  a.x  = rowp[k0];
                    a.y  = rowp[k0 + 1];
            v2f bm; bm.x = wm[k0 + koff];        // B[k,n] = Wm[k] for every column n
                    bm.y = wm[k0 + koff + 1];
            acc = __builtin_amdgcn_wmma_f32_16x16x4_f32(
                      false, a, false, bm, (short)0, acc, false, false);
        }
        // every column of C equals the row dot; VGPR j: lanes0-15 -> M=j, lanes16-31 -> M=8+j
        if (mrow == 0) {
            const int rbase = r0 + (lane >> 4) * 8;
#pragma unroll
            for (int j = 0; j < 8; ++j)
                g[rbase + j] = tanhf(acc[j] + ab);
        }
    }
    __syncthreads();

    // ---- Phase C: softmax over g[0..4095] ----
    float lmax = -3.402823466e38f;
    for (int i = tid; i < SLEN; i += BLOCK) lmax = fmaxf(lmax, g[i]);
    red[tid] = lmax;
    __syncthreads();
    for (int off = BLOCK / 2; off > 0; off >>= 1) {
        if (tid < off) red[tid] = fmaxf(red[tid], red[tid + off]);
        __syncthreads();
    }
    const float gmax = red[0];
    __syncthreads();

    float lsum = 0.0f;
    for (int i = tid; i < SLEN; i += BLOCK) {
        float e = __expf(g[i] - gmax);
        g[i] = e;
        lsum += e;
    }
    red[tid] = lsum;
    __syncthreads();
    for (int off = BLOCK / 2; off > 0; off >>= 1) {
        if (tid < off) red[tid] += red[tid + off];
        __syncthreads();
    }
    const float inv = 1.0f / red[0];
    __syncthreads();
    for (int i = tid; i < SLEN; i += BLOCK) g[i] *= inv;   // g is now alfa
    __syncthreads();

    // ---- Phase D: out[b, d0+n] = sum_s alfa[s] * mem[s, d0+n] via WMMA ----
    // wave w owns columns d0 = w*16. A[m,k] = alfa[s0+k] (same for all m);
    // B[k,n] = mem[s0+k, d0+n]. Every row of C is the same 16-wide partial sum.
    const int d0   = wave * 16;
    const int srow = (lane >> 4) << 1;           // 0 or 2
    const int ncol = lane & 15;
    v8f acc = {};
    for (int s0 = 0; s0 < SLEN; s0 += 4) {
        __builtin_prefetch(&mem[(size_t)(s0 + 8) * DDIM + d0 + ncol], 0, 0);
        v2f a;  a.x  = g[s0 + srow];
                a.y  = g[s0 + srow + 1];
        v2f bm; bm.x = mem[(size_t)(s0 + srow)     * DDIM + d0 + ncol];
                bm.y = mem[(size_t)(s0 + srow + 1) * DDIM + d0 + ncol];
        acc = __builtin_amdgcn_wmma_f32_16x16x4_f32(
                  false, a, false, bm, (short)0, acc, false, false);
    }
    if (lane < 16) out[b * DDIM + d0 + lane] = acc[0];     // C[0, lane]
}

extern "C" void kernel_launch(void* const* d_in, const int* in_sizes, int n_in,
                              void* d_out, int out_size, void* d_ws, size_t ws_size,
                              hipStream_t stream) {
    const float* aspect = (const float*)d_in[0];   // (B,1,D)
    const float* memory = (const float*)d_in[1];   // (B,S,D)
    const float* W      = (const float*)d_in[2];   // (2D,1)
    const float* bias   = (const float*)d_in[3];   // (1,)
    float* out = (float*)d_out;                    // (B,D)

    const int B = in_sizes[0] / DDIM;
    AttMemoryLayer_kernel<<<B, BLOCK, 0, stream>>>(aspect, memory, W, bias, out);
}